// PeptGAINET_4492535792516
// MI455X (gfx1250) — compile-verified
//
#include <hip/hip_runtime.h>
#include <hip/hip_bf16.h>

typedef __attribute__((ext_vector_type(2))) float v2f;
typedef __attribute__((ext_vector_type(8))) float v8f;

#define Bn   16
#define NPn  64
#define NTn  1024
#define NDn  64
#define Hh   4
#define Dd   32
#define ENCn 128
#define FUSEDn 256

__device__ __forceinline__ v8f wmma_f32(v2f a, v2f b, v8f c) {
  // D = A(16x4 f32) * B(4x16 f32) + C(16x16 f32)
  return __builtin_amdgcn_wmma_f32_16x16x4_f32(false, a, false, b, (short)0, c, false, false);
}

// ---------------------------------------------------------------------------
// C[M,N] = act(A[M,K] @ W[K,N] + bias)   (weights shared across batch)
// grid (M/16, N/64), block 32: one wave computes a 16x64 tile (4 accumulators)
// reusing the A fragment across 4 B fragments.
// ---------------------------------------------------------------------------
__global__ void gemm_rm(const float* __restrict__ A, const float* __restrict__ W,
                        const float* __restrict__ bias, float* __restrict__ C,
                        int M, int K, int N, int act) {
  const int m0 = blockIdx.x * 16, n0 = blockIdx.y * 64;
  const int lane = threadIdx.x, half = lane >> 4, lr = lane & 15;
  v8f acc0 = {}, acc1 = {}, acc2 = {}, acc3 = {};
  const float* Ap = A + (size_t)(m0 + lr) * K;
  const float* Wp = W + n0 + lr;
  for (int k0 = 0; k0 < K; k0 += 4) {
    const int ka = k0 + 2 * half;
    const v2f a = *(const v2f*)(Ap + ka);                 // A rows lr, K = ka, ka+1
    const float* w0 = Wp + (size_t)ka * N;
    const float* w1 = Wp + (size_t)(ka + 1) * N;
    v2f b;
    b.x = w0[0];  b.y = w1[0];  acc0 = wmma_f32(a, b, acc0);
    b.x = w0[16]; b.y = w1[16]; acc1 = wmma_f32(a, b, acc1);
    b.x = w0[32]; b.y = w1[32]; acc2 = wmma_f32(a, b, acc2);
    b.x = w0[48]; b.y = w1[48]; acc3 = wmma_f32(a, b, acc3);
  }
  v8f accs[4] = {acc0, acc1, acc2, acc3};
#pragma unroll
  for (int t = 0; t < 4; ++t) {
    const int col = n0 + t * 16 + lr;
    float bi = bias ? bias[col] : 0.f;
#pragma unroll
    for (int v = 0; v < 8; ++v) {
      const int row = m0 + v + 8 * half;
      float c = accs[t][v] + bi;
      if (act == 1) c = 1.0f / (1.0f + __expf(-c));
      C[(size_t)row * N + col] = c;
    }
  }
}

// ---------------------------------------------------------------------------
// src/dst attention coefficients: one thread per (b,n,h)
// ---------------------------------------------------------------------------
__global__ void attn_src_dst(const float* __restrict__ hfeat,
                             const float* __restrict__ a_src, const float* __restrict__ a_dst,
                             float* __restrict__ src, float* __restrict__ dst, int total) {
  const int idx = blockIdx.x * blockDim.x + threadIdx.x;
  if (idx >= total) return;
  const int h = idx & (Hh - 1);
  const float* hp = hfeat + (size_t)idx * Dd;
  float s = 0.f, t = 0.f;
#pragma unroll
  for (int d = 0; d < Dd; ++d) {
    const float v = hp[d];
    s += v * a_src[h * Dd + d];
    t += v * a_dst[h * Dd + d];
  }
  src[idx] = s; dst[idx] = t;
}

// ---------------------------------------------------------------------------
// Fused GAT attention softmax + WMMA aggregation.
// grid (Nn/16, H, B), block 32.  out[b,i,h*D+d] (pre-layernorm).
// The softmax A fragment is computed in registers (no alpha materialization)
// and reused across the two d-halves (D=32).
// ---------------------------------------------------------------------------
__global__ void gat_agg(const float* __restrict__ hfeat, const float* __restrict__ adj,
                        const float* __restrict__ mask, const float* __restrict__ src,
                        const float* __restrict__ dst, float* __restrict__ out, int Nn) {
  const int i0 = blockIdx.x * 16, head = blockIdx.y, b = blockIdx.z;
  const int lane = threadIdx.x, half = lane >> 4, lr = lane & 15;
  __shared__ float sl[16], smk[16];

  const int rowi = i0 + lr;
  const float src_i  = src[((size_t)b * Nn + rowi) * Hh + head];
  const float mask_i = mask[(size_t)b * Nn + rowi];
  const float* adjrow = adj + ((size_t)b * Nn + rowi) * Nn;
  const float* dstb   = dst + (size_t)b * Nn * Hh + head;
  const float* maskb  = mask + (size_t)b * Nn;

  // pass 1: row max (lanes L and L+16 split even/odd j of row lr)
  float m = -3.0e38f;
  for (int j = half; j < Nn; j += 2) {
    float e = src_i + dstb[(size_t)j * Hh];
    e = e > 0.f ? e : 0.2f * e;
    const bool valid = (adjrow[j] > 0.f) && (mask_i > 0.f) && (maskb[j] > 0.f);
    e = valid ? e : -1e9f;
    m = fmaxf(m, e);
  }
  m = fmaxf(m, __shfl_xor(m, 16, 32));
  // pass 2: sum of exp
  float l = 0.f;
  for (int j = half; j < Nn; j += 2) {
    float e = src_i + dstb[(size_t)j * Hh];
    e = e > 0.f ? e : 0.2f * e;
    const bool valid = (adjrow[j] > 0.f) && (mask_i > 0.f) && (maskb[j] > 0.f);
    e = valid ? e : -1e9f;
    l += __expf(e - m);
  }
  l += __shfl_xor(l, 16, 32);
  if (lane < 16) { sl[lr] = l; smk[lr] = mask_i; }
  __syncthreads();

  // pass 3: WMMA aggregation, A = exp(e - m) on the fly
  v8f acc0 = {}, acc1 = {};
  const float* hb = hfeat + (size_t)b * Nn * (Hh * Dd) + head * Dd;
  for (int j0 = 0; j0 < Nn; j0 += 4) {
    const int ja = j0 + 2 * half;
    float e0 = src_i + dstb[(size_t)ja * Hh];
    e0 = e0 > 0.f ? e0 : 0.2f * e0;
    e0 = ((adjrow[ja] > 0.f) && (mask_i > 0.f) && (maskb[ja] > 0.f)) ? e0 : -1e9f;
    float e1 = src_i + dstb[(size_t)(ja + 1) * Hh];
    e1 = e1 > 0.f ? e1 : 0.2f * e1;
    e1 = ((adjrow[ja + 1] > 0.f) && (mask_i > 0.f) && (maskb[ja + 1] > 0.f)) ? e1 : -1e9f;
    v2f a; a.x = __expf(e0 - m); a.y = __expf(e1 - m);
    const float* h0 = hb + (size_t)ja * (Hh * Dd) + lr;
    const float* h1 = hb + (size_t)(ja + 1) * (Hh * Dd) + lr;
    v2f b0, b1;
    b0.x = h0[0];  b0.y = h1[0];
    b1.x = h0[16]; b1.y = h1[16];
    acc0 = wmma_f32(a, b0, acc0);
    acc1 = wmma_f32(a, b1, acc1);
  }
#pragma unroll
  for (int v = 0; v < 8; ++v) {
    const int r = v + 8 * half;
    const float li = sl[r];
    const float sc = (li > 0.f ? 1.0f / li : 0.f) * smk[r];
    float* op = out + ((size_t)b * Nn + i0 + r) * (Hh * Dd) + head * Dd;
    op[lr]      = acc0[v] * sc;
    op[16 + lr] = acc1[v] * sc;
  }
}

// ---------------------------------------------------------------------------
// masked layernorm over ENC=128: one block (128 threads) per (b,i) row
// ---------------------------------------------------------------------------
__global__ void ln_mask(const float* __restrict__ x, const float* __restrict__ mask,
                        const float* __restrict__ g, const float* __restrict__ bta,
                        float* __restrict__ out) {
  const int row = blockIdx.x, t = threadIdx.x;
  const float mk = mask[row];
  const float v = x[(size_t)row * ENCn + t] * mk;
  __shared__ float red[ENCn];
  red[t] = v; __syncthreads();
  for (int s = 64; s > 0; s >>= 1) { if (t < s) red[t] += red[t + s]; __syncthreads(); }
  const float mean = red[0] * (1.0f / ENCn); __syncthreads();
  const float d = v - mean;
  red[t] = d * d; __syncthreads();
  for (int s = 64; s > 0; s >>= 1) { if (t < s) red[t] += red[t + s]; __syncthreads(); }
  const float var = red[0] * (1.0f / ENCn);
  out[(size_t)row * ENCn + t] = d * rsqrtf(var + 1e-5f) * g[t] + bta[t];
}

// ---------------------------------------------------------------------------
// Batched C[b,i,j] = scale * (A[b,i,:] . Bm[b,j,:]),  -1e9 where masked.
// grid (M/16, Nn/64, B), block 32: 16x64 tile per wave, A-frag reused x4,
// all fragment loads are contiguous 64-bit loads.
// ---------------------------------------------------------------------------
__global__ void gemm_bat_abt(const float* __restrict__ A, const float* __restrict__ Bm,
                             float* __restrict__ C, int M, int K, int Nn, float scale,
                             const float* __restrict__ rmask, const float* __restrict__ cmask) {
  const int m0 = blockIdx.x * 16, n0 = blockIdx.y * 64, b = blockIdx.z;
  const int lane = threadIdx.x, half = lane >> 4, lr = lane & 15;
  const float* Ap = A + (size_t)b * M * K + (size_t)(m0 + lr) * K;
  const float* Bp = Bm + (size_t)b * Nn * K + (size_t)(n0 + lr) * K;
  float* Cb = C + (size_t)b * M * Nn;
  v8f acc0 = {}, acc1 = {}, acc2 = {}, acc3 = {};
  for (int k0 = 0; k0 < K; k0 += 4) {
    const int ka = k0 + 2 * half;
    const v2f a  = *(const v2f*)(Ap + ka);
    const v2f f0 = *(const v2f*)(Bp + ka);
    const v2f f1 = *(const v2f*)(Bp + (size_t)16 * K + ka);
    const v2f f2 = *(const v2f*)(Bp + (size_t)32 * K + ka);
    const v2f f3 = *(const v2f*)(Bp + (size_t)48 * K + ka);
    acc0 = wmma_f32(a, f0, acc0);
    acc1 = wmma_f32(a, f1, acc1);
    acc2 = wmma_f32(a, f2, acc2);
    acc3 = wmma_f32(a, f3, acc3);
  }
  v8f accs[4] = {acc0, acc1, acc2, acc3};
#pragma unroll
  for (int t = 0; t < 4; ++t) {
    const int col = n0 + t * 16 + lr;
    const bool cv = cmask[(size_t)b * Nn + col] > 0.f;
#pragma unroll
    for (int v = 0; v < 8; ++v) {
      const int row = m0 + v + 8 * half;
      const bool valid = cv && (rmask[(size_t)b * M + row] > 0.f);
      Cb[(size_t)row * Nn + col] = valid ? accs[t][v] * scale : -1e9f;
    }
  }
}

// ---------------------------------------------------------------------------
// Batched C[b,i,coff+n] = A[b,i,:] @ Bm[b,:,n]  (C leading dim ldC)
// grid (M/16, Nn/64, B), block 32: 16x64 tile per wave.
// ---------------------------------------------------------------------------
__global__ void gemm_bat_ab(const float* __restrict__ A, const float* __restrict__ Bm,
                            float* __restrict__ C, int M, int K, int Nn, int ldC, int coff) {
  const int m0 = blockIdx.x * 16, n0 = blockIdx.y * 64, b = blockIdx.z;
  const int lane = threadIdx.x, half = lane >> 4, lr = lane & 15;
  const float* Ap = A + (size_t)b * M * K + (size_t)(m0 + lr) * K;
  const float* Bp = Bm + (size_t)b * K * Nn + n0 + lr;
  float* Cb = C + (size_t)b * M * ldC;
  v8f acc0 = {}, acc1 = {}, acc2 = {}, acc3 = {};
  for (int k0 = 0; k0 < K; k0 += 4) {
    const int ka = k0 + 2 * half;
    const v2f a = *(const v2f*)(Ap + ka);
    const float* b0 = Bp + (size_t)ka * Nn;
    const float* b1 = Bp + (size_t)(ka + 1) * Nn;
    v2f f;
    f.x = b0[0];  f.y = b1[0];  acc0 = wmma_f32(a, f, acc0);
    f.x = b0[16]; f.y = b1[16]; acc1 = wmma_f32(a, f, acc1);
    f.x = b0[32]; f.y = b1[32]; acc2 = wmma_f32(a, f, acc2);
    f.x = b0[48]; f.y = b1[48]; acc3 = wmma_f32(a, f, acc3);
  }
  v8f accs[4] = {acc0, acc1, acc2, acc3};
#pragma unroll
  for (int t = 0; t < 4; ++t) {
    const int col = coff + n0 + t * 16 + lr;
#pragma unroll
    for (int v = 0; v < 8; ++v) {
      const int row = m0 + v + 8 * half;
      Cb[(size_t)row * ldC + col] = accs[t][v];
    }
  }
}

// softmax over last dim, one 256-thread block per row
__global__ void softmax_row(const float* __restrict__ x, float* __restrict__ y, int C) {
  const int row = blockIdx.x, t = threadIdx.x;
  const float* xr = x + (size_t)row * C;
  float* yr = y + (size_t)row * C;
  __shared__ float red[256];
  float m = -3.0e38f;
  for (int c = t; c < C; c += 256) m = fmaxf(m, xr[c]);
  red[t] = m; __syncthreads();
  for (int s = 128; s > 0; s >>= 1) { if (t < s) red[t] = fmaxf(red[t], red[t + s]); __syncthreads(); }
  m = red[0]; __syncthreads();
  float l = 0.f;
  for (int c = t; c < C; c += 256) l += __expf(xr[c] - m);
  red[t] = l; __syncthreads();
  for (int s = 128; s > 0; s >>= 1) { if (t < s) red[t] += red[t + s]; __syncthreads(); }
  const float inv = 1.0f / red[0];
  for (int c = t; c < C; c += 256) yr[c] = __expf(xr[c] - m) * inv;
}

// transposed softmax: t2p[b,j,i] = softmax_i(score[b,i,j]); one 64-thread block per (b,j)
__global__ void softmax_t2p(const float* __restrict__ score, float* __restrict__ t2p) {
  const int row = blockIdx.x;               // b*NT + j
  const int b = row / NTn, j = row % NTn, t = threadIdx.x;
  const float v = score[(size_t)b * NPn * NTn + (size_t)t * NTn + j];
  __shared__ float red[64];
  red[t] = v; __syncthreads();
  for (int s = 32; s > 0; s >>= 1) { if (t < s) red[t] = fmaxf(red[t], red[t + s]); __syncthreads(); }
  const float m = red[0]; __syncthreads();
  const float e = __expf(v - m);
  red[t] = e; __syncthreads();
  for (int s = 32; s > 0; s >>= 1) { if (t < s) red[t] += red[t + s]; __syncthreads(); }
  t2p[(size_t)row * NPn + t] = e / red[0];
}

__global__ void copy_cols(const float* __restrict__ src, float* __restrict__ dst,
                          int C, int ldD, int coff, size_t total) {
  const size_t idx = blockIdx.x * (size_t)blockDim.x + threadIdx.x;
  if (idx >= total) return;
  const size_t r = idx / C;
  const int c = (int)(idx % C);
  dst[r * ldD + coff + c] = src[idx];
}

__global__ void mul_inplace(float* __restrict__ x, const float* __restrict__ g, size_t n) {
  const size_t idx = blockIdx.x * (size_t)blockDim.x + threadIdx.x;
  if (idx < n) x[idx] *= g[idx];
}

// masked mean over node axis: one thread per (b, c)
__global__ void masked_mean(const float* __restrict__ x, const float* __restrict__ mask,
                            float* __restrict__ out, int Nn, int C, int total) {
  const int idx = blockIdx.x * blockDim.x + threadIdx.x;
  if (idx >= total) return;
  const int b = idx / C, c = idx % C;
  const float* xb = x + (size_t)b * Nn * C + c;
  const float* mb = mask + (size_t)b * Nn;
  float s = 0.f, ms = 0.f;
  for (int i = 0; i < Nn; ++i) { const float m = mb[i]; s += xb[(size_t)i * C] * m; ms += m; }
  out[idx] = s / fmaxf(ms, 1.0f);
}

// RESCAL + MLP head: one 256-thread block per batch element
__global__ void head_kernel(const float* __restrict__ pep_vec, const float* __restrict__ prot_vec,
                            const float* __restrict__ rel, const float* __restrict__ w1,
                            const float* __restrict__ b1, const float* __restrict__ w2,
                            const float* __restrict__ b2, float* __restrict__ out) {
  const int b = blockIdx.x, t = threadIdx.x;
  __shared__ float p[FUSEDn], q[FUSEDn], mix[4 * FUSEDn], red[FUSEDn];
  p[t] = pep_vec[b * FUSEDn + t];
  q[t] = prot_vec[b * FUSEDn + t];
  __syncthreads();
  float r = 0.f;
  for (int k = 0; k < FUSEDn; ++k) r += p[k] * rel[k * FUSEDn + t];
  red[t] = r * q[t]; __syncthreads();
  for (int s = 128; s > 0; s >>= 1) { if (t < s) red[t] += red[t + s]; __syncthreads(); }
  const float bil = red[0]; __syncthreads();
  mix[t] = p[t]; mix[FUSEDn + t] = q[t];
  mix[2 * FUSEDn + t] = fabsf(p[t] - q[t]); mix[3 * FUSEDn + t] = p[t] * q[t];
  __syncthreads();
  float h = 0.f;
  for (int k = 0; k < 4 * FUSEDn; ++k) h += mix[k] * w1[k * FUSEDn + t];
  h = fmaxf(h + b1[t], 0.f);
  red[t] = h * w2[t]; __syncthreads();
  for (int s = 128; s > 0; s >>= 1) { if (t < s) red[t] += red[t + s]; __syncthreads(); }
  if (t == 0) {
    const float logit = bil + red[0] + b2[0];
    out[b] = logit;
    out[Bn + b] = 1.0f / (1.0f + __expf(-logit));
  }
}

// ---------------------------------------------------------------------------
extern "C" void kernel_launch(void* const* d_in, const int* in_sizes, int n_in,
                              void* d_out, int out_size, void* d_ws, size_t ws_size,
                              hipStream_t stream) {
  (void)in_sizes; (void)n_in; (void)out_size; (void)ws_size;
  const float* pep_x    = (const float*)d_in[0];
  const float* pep_adj  = (const float*)d_in[1];
  const float* pep_mask = (const float*)d_in[2];
  const float* prot_x    = (const float*)d_in[3];
  const float* prot_adj  = (const float*)d_in[4];
  const float* prot_mask = (const float*)d_in[5];
  const float* pep0_w = (const float*)d_in[6];  const float* pep0_as = (const float*)d_in[7];
  const float* pep0_ad = (const float*)d_in[8]; const float* pep0_g = (const float*)d_in[9];
  const float* pep0_b = (const float*)d_in[10];
  const float* pep1_w = (const float*)d_in[11]; const float* pep1_as = (const float*)d_in[12];
  const float* pep1_ad = (const float*)d_in[13]; const float* pep1_g = (const float*)d_in[14];
  const float* pep1_b = (const float*)d_in[15];
  const float* prot0_w = (const float*)d_in[16]; const float* prot0_as = (const float*)d_in[17];
  const float* prot0_ad = (const float*)d_in[18]; const float* prot0_g = (const float*)d_in[19];
  const float* prot0_b = (const float*)d_in[20];
  const float* prot1_w = (const float*)d_in[21]; const float* prot1_as = (const float*)d_in[22];
  const float* prot1_ad = (const float*)d_in[23]; const float* prot1_g = (const float*)d_in[24];
  const float* prot1_b = (const float*)d_in[25];
  const float* ca_wp = (const float*)d_in[26]; const float* ca_wt = (const float*)d_in[27];
  const float* gate_pw = (const float*)d_in[28]; const float* gate_pb = (const float*)d_in[29];
  const float* gate_tw = (const float*)d_in[30]; const float* gate_tb = (const float*)d_in[31];
  const float* rel = (const float*)d_in[32];
  const float* mlp_w1 = (const float*)d_in[33]; const float* mlp_b1 = (const float*)d_in[34];
  const float* mlp_w2 = (const float*)d_in[35]; const float* mlp_b2 = (const float*)d_in[36];

  float* ws = (float*)d_ws;
  size_t o = 0;
  auto alloc = [&](size_t n) { float* p = ws + o; o += n; return p; };
  float* pep_h0  = alloc((size_t)Bn * NPn * ENCn);     // 131072
  float* pep_h1  = alloc((size_t)Bn * NPn * ENCn);
  float* pep_tmp = alloc((size_t)Bn * NPn * ENCn);
  float* qb      = alloc((size_t)Bn * NPn * ENCn);
  float* pep2    = alloc((size_t)Bn * NPn * FUSEDn);   // 262144
  float* srcb    = alloc((size_t)Bn * NTn * Hh);       // 65536 (sized for protein)
  float* dstb    = alloc((size_t)Bn * NTn * Hh);
  float* pep_vecb  = alloc((size_t)Bn * FUSEDn);
  float* prot_vecb = alloc((size_t)Bn * FUSEDn);
  float* score  = alloc((size_t)Bn * NPn * NTn);       // 1048576
  float* p2t    = alloc((size_t)Bn * NPn * NTn);
  float* t2p    = alloc((size_t)Bn * NTn * NPn);
  float* prot_h0  = alloc((size_t)Bn * NTn * ENCn);    // 2097152
  float* prot_h1  = alloc((size_t)Bn * NTn * ENCn);
  float* prot_tmp = alloc((size_t)Bn * NTn * ENCn);
  float* prot2    = alloc((size_t)Bn * NTn * FUSEDn);  // 4194304
  // aliases (regions dead by the time they are reused):
  float* kb     = prot_tmp;   // k projection, after protein layers done with prot_tmp
  float* gate_p = score;      // score consumed by both softmaxes before gating
  float* gate_t = prot_h0;    // spans prot_h0+prot_h1 (both dead after ctx/concat)

  const dim3 w32(32);

  // ---- Peptide GAT layer 0 ----
  gemm_rm<<<dim3(Bn * NPn / 16, ENCn / 64), w32, 0, stream>>>(pep_x, pep0_w, nullptr, pep_h0, Bn * NPn, NDn, ENCn, 0);
  attn_src_dst<<<(Bn * NPn * Hh + 127) / 128, 128, 0, stream>>>(pep_h0, pep0_as, pep0_ad, srcb, dstb, Bn * NPn * Hh);
  gat_agg<<<dim3(NPn / 16, Hh, Bn), w32, 0, stream>>>(pep_h0, pep_adj, pep_mask, srcb, dstb, pep_tmp, NPn);
  ln_mask<<<Bn * NPn, ENCn, 0, stream>>>(pep_tmp, pep_mask, pep0_g, pep0_b, pep_h1);
  // ---- Peptide GAT layer 1 ----
  gemm_rm<<<dim3(Bn * NPn / 16, ENCn / 64), w32, 0, stream>>>(pep_h1, pep1_w, nullptr, pep_h0, Bn * NPn, ENCn, ENCn, 0);
  attn_src_dst<<<(Bn * NPn * Hh + 127) / 128, 128, 0, stream>>>(pep_h0, pep1_as, pep1_ad, srcb, dstb, Bn * NPn * Hh);
  gat_agg<<<dim3(NPn / 16, Hh, Bn), w32, 0, stream>>>(pep_h0, pep_adj, pep_mask, srcb, dstb, pep_tmp, NPn);
  ln_mask<<<Bn * NPn, ENCn, 0, stream>>>(pep_tmp, pep_mask, pep1_g, pep1_b, pep_h1);

  // ---- Protein GAT layer 0 ----
  gemm_rm<<<dim3(Bn * NTn / 16, ENCn / 64), w32, 0, stream>>>(prot_x, prot0_w, nullptr, prot_h0, Bn * NTn, NDn, ENCn, 0);
  attn_src_dst<<<(Bn * NTn * Hh + 127) / 128, 128, 0, stream>>>(prot_h0, prot0_as, prot0_ad, srcb, dstb, Bn * NTn * Hh);
  gat_agg<<<dim3(NTn / 16, Hh, Bn), w32, 0, stream>>>(prot_h0, prot_adj, prot_mask, srcb, dstb, prot_tmp, NTn);
  ln_mask<<<Bn * NTn, ENCn, 0, stream>>>(prot_tmp, prot_mask, prot0_g, prot0_b, prot_h1);
  // ---- Protein GAT layer 1 ----
  gemm_rm<<<dim3(Bn * NTn / 16, ENCn / 64), w32, 0, stream>>>(prot_h1, prot1_w, nullptr, prot_h0, Bn * NTn, ENCn, ENCn, 0);
  attn_src_dst<<<(Bn * NTn * Hh + 127) / 128, 128, 0, stream>>>(prot_h0, prot1_as, prot1_ad, srcb, dstb, Bn * NTn * Hh);
  gat_agg<<<dim3(NTn / 16, Hh, Bn), w32, 0, stream>>>(prot_h0, prot_adj, prot_mask, srcb, dstb, prot_tmp, NTn);
  ln_mask<<<Bn * NTn, ENCn, 0, stream>>>(prot_tmp, prot_mask, prot1_g, prot1_b, prot_h1);

  // ---- Co-attention ----
  gemm_rm<<<dim3(Bn * NPn / 16, ENCn / 64), w32, 0, stream>>>(pep_h1, ca_wp, nullptr, qb, Bn * NPn, ENCn, ENCn, 0);
  gemm_rm<<<dim3(Bn * NTn / 16, ENCn / 64), w32, 0, stream>>>(prot_h1, ca_wt, nullptr, kb, Bn * NTn, ENCn, ENCn, 0);
  gemm_bat_abt<<<dim3(NPn / 16, NTn / 64, Bn), w32, 0, stream>>>(qb, kb, score, NPn, ENCn, NTn,
                                                                 1.0f / sqrtf((float)ENCn), pep_mask, prot_mask);
  softmax_row<<<Bn * NPn, 256, 0, stream>>>(score, p2t, NTn);
  softmax_t2p<<<Bn * NTn, 64, 0, stream>>>(score, t2p);
  // contexts written straight into the second half of the concat buffers
  gemm_bat_ab<<<dim3(NPn / 16, ENCn / 64, Bn), w32, 0, stream>>>(p2t, prot_h1, pep2, NPn, NTn, ENCn, FUSEDn, ENCn);
  gemm_bat_ab<<<dim3(NTn / 16, ENCn / 64, Bn), w32, 0, stream>>>(t2p, pep_h1, prot2, NTn, NPn, ENCn, FUSEDn, ENCn);
  {
    size_t tp = (size_t)Bn * NPn * ENCn;
    copy_cols<<<(unsigned)((tp + 255) / 256), 256, 0, stream>>>(pep_h1, pep2, ENCn, FUSEDn, 0, tp);
    size_t tt = (size_t)Bn * NTn * ENCn;
    copy_cols<<<(unsigned)((tt + 255) / 256), 256, 0, stream>>>(prot_h1, prot2, ENCn, FUSEDn, 0, tt);
  }

  // ---- Gated fusion ----
  gemm_rm<<<dim3(Bn * NPn / 16, FUSEDn / 64), w32, 0, stream>>>(pep2, gate_pw, gate_pb, gate_p, Bn * NPn, FUSEDn, FUSEDn, 1);
  mul_inplace<<<(unsigned)(((size_t)Bn * NPn * FUSEDn + 255) / 256), 256, 0, stream>>>(pep2, gate_p, (size_t)Bn * NPn * FUSEDn);
  gemm_rm<<<dim3(Bn * NTn / 16, FUSEDn / 64), w32, 0, stream>>>(prot2, gate_tw, gate_tb, gate_t, Bn * NTn, FUSEDn, FUSEDn, 1);
  mul_inplace<<<(unsigned)(((size_t)Bn * NTn * FUSEDn + 255) / 256), 256, 0, stream>>>(prot2, gate_t, (size_t)Bn * NTn * FUSEDn);

  // ---- Pooling + head ----
  masked_mean<<<(Bn * FUSEDn + 255) / 256, 256, 0, stream>>>(pep2, pep_mask, pep_vecb, NPn, FUSEDn, Bn * FUSEDn);
  masked_mean<<<(Bn * FUSEDn + 255) / 256, 256, 0, stream>>>(prot2, prot_mask, prot_vecb, NTn, FUSEDn, Bn * FUSEDn);
  head_kernel<<<Bn, FUSEDn, 0, stream>>>(pep_vecb, prot_vecb, rel, mlp_w1, mlp_b1, mlp_w2, mlp_b2, (float*)d_out);
}